// cTanh_34007551049882
// MI455X (gfx1250) — compile-verified
//
#include <hip/hip_runtime.h>
#include <stdint.h>

// CDNA5 / gfx1250: wave32, WGP. This workload is pure streaming elementwise
// (512 MiB @ 23.3 TB/s ~= 23 us floor), so the kernel is built around B128
// non-temporal global traffic + an LDS-resident segment table staged with the
// gfx1250 async global->LDS DMA path (ASYNCcnt).

typedef float floatx4 __attribute__((ext_vector_type(4)));
typedef int   v4i     __attribute__((vector_size(16)));   // matches builtin proto

__device__ __forceinline__ float pwl_eval(float v, const float2* __restrict__ sab,
                                          float x0, float inv_dx,
                                          float x62, float x63) {
    // Uniform-grid segment index (replaces searchsorted): segment i covers
    // [x[i], x[i+1]); clamp into [0, 62].
    float t  = (v - x0) * inv_dx;
    int  idx = (int)t;
    idx = idx < 0  ? 0  : idx;
    idx = idx > 62 ? 62 : idx;
    float2 ab  = sab[idx];                       // ds_load_b64 gather
    float  seg = __builtin_fmaf(ab.x, v, ab.y);  // a*v + b
    float  f   = ((v >= x0) && (v < x63)) ? seg : 0.0f;
    f += (v >= x62) ? 1.0f : 0.0f;               // reference's preserved bug
    f -= (v <  x0)  ? 1.0f : 0.0f;               // -1 below range
    return f;
}

__global__ __launch_bounds__(256) void
pwl_tanh_kernel(const float* __restrict__ in, const float* __restrict__ tab,
                float* __restrict__ out, int n) {
    __shared__ float  s_raw[128];   // raw table: 64 x {x_raw, y_raw} = 512 B
    __shared__ float2 s_ab[64];     // per-segment {slope a, offset b}

    const int t = threadIdx.x;

    // ---- Stage the 512 B table into LDS via gfx1250 async DMA (ASYNCcnt) ----
#if __has_builtin(__builtin_amdgcn_global_load_async_to_lds_b128)
    if (t < 32) {  // wave 0 fully active: 32 lanes x B128 = 512 B
        __builtin_amdgcn_global_load_async_to_lds_b128(
            (__attribute__((address_space(1))) v4i*)(tab + t * 4),
            (__attribute__((address_space(3))) v4i*)(s_raw + t * 4),
            0, 0);
#if __has_builtin(__builtin_amdgcn_s_wait_asynccnt)
        __builtin_amdgcn_s_wait_asynccnt(0);
#else
        asm volatile("s_wait_asynccnt 0" ::: "memory");
#endif
    }
#else
    if (t < 128) s_raw[t] = tab[t];
#endif
    __syncthreads();

    // ---- Build per-segment slope/offset in LDS (63 segments) ----
    if (t < 63) {
        float xi = s_raw[2 * t]              * 10.0f;
        float yi = (s_raw[2 * t + 1] - 0.9f) * 10.0f;
        float xj = s_raw[2 * t + 2]          * 10.0f;
        float yj = (s_raw[2 * t + 3] - 0.9f) * 10.0f;
        float a  = (yj - yi) / (xj - xi);
        s_ab[t]  = make_float2(a, __builtin_fmaf(-a, xi, yi));  // b = y - a*x
    }
    __syncthreads();

    const float x0     = s_raw[0]   * 10.0f;   // LDS broadcast reads
    const float x62    = s_raw[124] * 10.0f;
    const float x63    = s_raw[126] * 10.0f;
    const float inv_dx = 63.0f / (x63 - x0);

    // ---- Streaming main loop: B128 non-temporal load/store per lane ----
    const int       n4     = n >> 2;
    const floatx4*  in4    = (const floatx4*)in;
    floatx4*        out4   = (floatx4*)out;
    const int       stride = (int)(gridDim.x * blockDim.x);

    for (int i = (int)(blockIdx.x * blockDim.x) + t; i < n4; i += stride) {
        floatx4 v = __builtin_nontemporal_load(in4 + i);
        floatx4 r;
        r.x = pwl_eval(v.x, s_ab, x0, inv_dx, x62, x63);
        r.y = pwl_eval(v.y, s_ab, x0, inv_dx, x62, x63);
        r.z = pwl_eval(v.z, s_ab, x0, inv_dx, x62, x63);
        r.w = pwl_eval(v.w, s_ab, x0, inv_dx, x62, x63);
        __builtin_nontemporal_store(r, out4 + i);
    }

    // Scalar tail (n not multiple of 4; unused for 2^26 but kept for safety)
    for (int i = (n4 << 2) + (int)(blockIdx.x * blockDim.x) + t; i < n; i += stride) {
        out[i] = pwl_eval(in[i], s_ab, x0, inv_dx, x62, x63);
    }
}

extern "C" void kernel_launch(void* const* d_in, const int* in_sizes, int n_in,
                              void* d_out, int out_size, void* d_ws, size_t ws_size,
                              hipStream_t stream) {
    (void)in_sizes; (void)n_in; (void)d_ws; (void)ws_size;
    const float* in  = (const float*)d_in[0];   // (32,4096,512) fp32
    const float* tab = (const float*)d_in[1];   // (64,2) fp32
    float*       out = (float*)d_out;

    const int n       = out_size;               // 67,108,864
    const int threads = 256;                    // 8 wave32 waves per block
    int blocks = (n / 4 + threads - 1) / threads;
    if (blocks > 8192) blocks = 8192;           // ~8 float4 iters/thread
    if (blocks < 1)    blocks = 1;

    pwl_tanh_kernel<<<blocks, threads, 0, stream>>>(in, tab, out, n);
}